// PixelContrastiveLoss_47364899340365
// MI455X (gfx1250) — compile-verified
//
#include <hip/hip_runtime.h>

typedef float v2f __attribute__((ext_vector_type(2)));
typedef float v8f __attribute__((ext_vector_type(8)));

#define M_TOT   8192        // 2*64*64 pixels
#define C_DIM   256         // channels
#define HW      4096        // 64*64
#define TILE    128         // logits tile per block
#define KC      64          // K chunk staged in LDS
#define LDSTR   68          // LDS row stride (floats): bank = 4*row + k -> conflict-free b64 frags
#define INV_T   10.0f       // 1 / temperature

// ---------------------------------------------------------------------------
// 1) (N,C,H,W) -> (M,C) transpose + L2 row normalization (torch F.normalize)
//    One thread per pixel m; C-loop loads are fully coalesced across the wave.
// ---------------------------------------------------------------------------
__global__ __launch_bounds__(256) void normalize_kernel(
    const float* __restrict__ feat, float* __restrict__ fnorm) {
  const int m  = blockIdx.x * 256 + threadIdx.x;
  const int n  = m >> 12;            // m / 4096
  const int hw = m & (HW - 1);
  const float* base = feat + (size_t)n * C_DIM * HW + hw;

  float ss = 0.0f;
  for (int c = 0; c < C_DIM; ++c) {
    float x = base[(size_t)c * HW];
    ss += x * x;
  }
  const float inv = 1.0f / fmaxf(sqrtf(ss), 1e-12f);

  float4* dst = (float4*)(fnorm + (size_t)m * C_DIM);
  for (int c = 0; c < C_DIM; c += 4) {
    float4 v;
    v.x = base[(size_t)(c + 0) * HW] * inv;
    v.y = base[(size_t)(c + 1) * HW] * inv;
    v.z = base[(size_t)(c + 2) * HW] * inv;
    v.w = base[(size_t)(c + 3) * HW] * inv;
    dst[c >> 2] = v;
  }
}

// ---------------------------------------------------------------------------
// 2) zero the row accumulators (harness does not re-zero workspace)
// ---------------------------------------------------------------------------
__global__ void zero_kernel(float* __restrict__ p) {
  p[blockIdx.x * 256 + threadIdx.x] = 0.0f;
}

// ---------------------------------------------------------------------------
// 3) Fused Gram-tile + masked exp row reduction.
//    Block = 256 threads (8 waves), computes a 128x128 tile of logits with
//    V_WMMA_F32_16X16X4_F32; wave w owns rows [i0+16w, +16), all 128 cols.
// ---------------------------------------------------------------------------
__global__ __launch_bounds__(256, 2) void gram_kernel(
    const float* __restrict__ fnorm, const long long* __restrict__ lab,
    float* __restrict__ rowExp, float* __restrict__ rowLin) {
  __shared__ float lA[TILE * LDSTR];
  __shared__ float lB[TILE * LDSTR];

  const int tid  = threadIdx.x;
  const int wave = tid >> 5;
  const int lane = tid & 31;
  const int i0   = blockIdx.y * TILE;
  const int j0   = blockIdx.x * TILE;

  v8f acc[8] = {};

  const int mrow = (wave << 4) + (lane & 15);      // A row within tile
  const int koff = (lane >> 4) << 1;               // lanes 16-31 hold K+2,K+3

  for (int kc = 0; kc < C_DIM; kc += KC) {
    // cooperative global->LDS stage: 128 rows x 64 K, float4 per access
#pragma unroll
    for (int it = 0; it < 8; ++it) {
      const int v   = tid + it * 256;              // 0..2047
      const int row = v >> 4;
      const int kk  = (v & 15) << 2;
      const float4 da = *(const float4*)&fnorm[(size_t)(i0 + row) * C_DIM + kc + kk];
      const float4 db = *(const float4*)&fnorm[(size_t)(j0 + row) * C_DIM + kc + kk];
      *(float4*)&lA[row * LDSTR + kk] = da;
      *(float4*)&lB[row * LDSTR + kk] = db;
    }
    __syncthreads();

    for (int k = 0; k < KC; k += 4) {
      const v2f a = *(const v2f*)&lA[mrow * LDSTR + k + koff];
#pragma unroll
      for (int t = 0; t < 8; ++t) {
        const v2f b = *(const v2f*)&lB[((t << 4) + (lane & 15)) * LDSTR + k + koff];
        acc[t] = __builtin_amdgcn_wmma_f32_16x16x4_f32(
            /*neg_a=*/false, a, /*neg_b=*/false, b,
            /*c_mod=*/(short)0, acc[t], /*reuse_a=*/false, /*reuse_b=*/false);
      }
    }
    __syncthreads();
  }

  // epilogue: logits = acc * 10; accumulate per-row sum(exp*mask) and sum(logit)
  long long lj[8];
#pragma unroll
  for (int t = 0; t < 8; ++t) lj[t] = lab[j0 + (t << 4) + (lane & 15)];

#pragma unroll
  for (int r = 0; r < 8; ++r) {
    // D layout: VGPR r holds row r (lanes 0-15) and row r+8 (lanes 16-31)
    const int irow = i0 + (wave << 4) + r + ((lane >> 4) << 3);
    const long long li = lab[irow];
    float se = 0.0f, sl = 0.0f;
#pragma unroll
    for (int t = 0; t < 8; ++t) {
      const float logit = acc[t][r] * INV_T;
      se += (li == lj[t]) ? __expf(logit) : 0.0f;
      sl += logit;
    }
    // reduce across the 16 column lanes of each half-wave
#pragma unroll
    for (int off = 1; off < 16; off <<= 1) {
      se += __shfl_xor(se, off, 32);
      sl += __shfl_xor(sl, off, 32);
    }
    if ((lane & 15) == 0) {
      unsafeAtomicAdd(&rowExp[irow], se);
      unsafeAtomicAdd(&rowLin[irow], sl);
    }
  }
}

// ---------------------------------------------------------------------------
// 4) loss = mean_i log(rowExp_i + 1e-6) - (1/M^2) * sum_i rowLin_i
// ---------------------------------------------------------------------------
__global__ __launch_bounds__(256) void finalize_kernel(
    const float* __restrict__ rowExp, const float* __restrict__ rowLin,
    float* __restrict__ out) {
  __shared__ float sE[256];
  __shared__ float sL[256];
  float e = 0.0f, l = 0.0f;
  for (int i = threadIdx.x; i < M_TOT; i += 256) {
    e += __logf(rowExp[i] + 1e-6f);
    l += rowLin[i];
  }
  sE[threadIdx.x] = e;
  sL[threadIdx.x] = l;
  __syncthreads();
  for (int s = 128; s > 0; s >>= 1) {
    if (threadIdx.x < s) {
      sE[threadIdx.x] += sE[threadIdx.x + s];
      sL[threadIdx.x] += sL[threadIdx.x + s];
    }
    __syncthreads();
  }
  if (threadIdx.x == 0)
    out[0] = sE[0] / (float)M_TOT - sL[0] / ((float)M_TOT * (float)M_TOT);
}

extern "C" void kernel_launch(void* const* d_in, const int* in_sizes, int n_in,
                              void* d_out, int out_size, void* d_ws, size_t ws_size,
                              hipStream_t stream) {
  const float*     feat = (const float*)d_in[0];
  const long long* lab  = (const long long*)d_in[1];
  float*           out  = (float*)d_out;

  float* fnorm = (float*)d_ws;                       // M x C fp32 (8 MB)
  float* rowE  = fnorm + (size_t)M_TOT * C_DIM;      // 8192 floats
  float* rowL  = rowE + M_TOT;                       // 8192 floats

  normalize_kernel<<<M_TOT / 256, 256, 0, stream>>>(feat, fnorm);
  zero_kernel<<<(2 * M_TOT) / 256, 256, 0, stream>>>(rowE);   // rowE+rowL contiguous
  dim3 grid(M_TOT / TILE, M_TOT / TILE);
  gram_kernel<<<grid, 256, 0, stream>>>(fnorm, lab, rowE, rowL);
  finalize_kernel<<<1, 256, 0, stream>>>(rowE, rowL, out);
}